// ShapeCarver_79413945303604
// MI455X (gfx1250) — compile-verified
//
#include <hip/hip_runtime.h>

typedef __attribute__((ext_vector_type(2))) float v2f;
typedef __attribute__((ext_vector_type(8))) float v8f;

#define NCAM   6
#define H      512
#define W      512
#define GS     96
#define NPTS   (GS*GS*GS)      // 884736
#define HWPIX  (H*W)           // 262144
#define FILLV  0.45f
#define NVW    0.25f

__device__ __forceinline__ int clampi(int v, int lo, int hi) {
    return v < lo ? lo : (v > hi ? hi : v);
}

// ---------------------------------------------------------------------------
// Kernel 1: build the 48x4 fused projection matrix (A-tiles for WMMA).
// Rows 3c..3c+2     : K_c * E_c[0:3] * T   (pixel rows)
// Rows 18+3c..+2    : E_c[0:3] * T         (camera-space rows, for distance)
// Rows 36..47       : zero padding
// T folds the z-rotation (angle) and center translation of the voxel grid.
// Stored row-major [48][4] == WMMA A fragment order (tile t, lane m = row%16,
// half h = K-pair), since a.xy = A[16t+m][2h..2h+1] -> linear index 4*row+k.
// ---------------------------------------------------------------------------
__global__ void k_setup(const float* __restrict__ center,
                        const float* __restrict__ angle,
                        const float* __restrict__ Km,
                        const float* __restrict__ Em,
                        float* __restrict__ wsA) {
    if (threadIdx.x != 0 || blockIdx.x != 0) return;
    for (int i = 0; i < 192; ++i) wsA[i] = 0.0f;
    float a  = angle[0];
    float ca = cosf(a), sa = sinf(a);
    float T[4][4] = {{ca, -sa, 0.0f, center[0]},
                     {sa,  ca, 0.0f, center[1]},
                     {0.0f, 0.0f, 1.0f, center[2]},
                     {0.0f, 0.0f, 0.0f, 1.0f}};
    for (int c = 0; c < NCAM; ++c) {
        float ET[3][4];
        for (int i = 0; i < 3; ++i)
            for (int j = 0; j < 4; ++j) {
                float s = 0.0f;
                for (int k = 0; k < 4; ++k) s += Em[c*16 + i*4 + k] * T[k][j];
                ET[i][j] = s;
            }
        float M[3][4];
        for (int i = 0; i < 3; ++i)
            for (int j = 0; j < 4; ++j) {
                float s = 0.0f;
                for (int k = 0; k < 3; ++k) s += Km[c*9 + i*3 + k] * ET[k][j];
                M[i][j] = s;
            }
        for (int i = 0; i < 3; ++i)
            for (int j = 0; j < 4; ++j) {
                wsA[(3*c + i)*4 + j]      = M[i][j];
                wsA[(18 + 3*c + i)*4 + j] = ET[i][j];
            }
    }
}

// ---------------------------------------------------------------------------
// Kernel 2: WMMA projection. Each wave (32 lanes) owns 16 points.
// B fragment (4x16, K=homogeneous coord): lanes 0-15 hold K=0,1 (gx,gy),
// lanes 16-31 hold K=2,3 (gz,1) for column n = lane&15.
// Three V_WMMA_F32_16X16X4_F32 produce all 48 rows; spill to LDS, then each
// lane processes 3 (camera,point) pairs: mask sample (z+1e-8), pixel id with
// clamped z, camera-space distance. Writes dist/pid/mask-count to workspace.
// ---------------------------------------------------------------------------
__global__ __launch_bounds__(256)
void k_project(const float* __restrict__ grid,
               const float* __restrict__ maskImg,
               const float* __restrict__ wsA,
               float* __restrict__ distbuf,
               unsigned int* __restrict__ pidbuf,
               float* __restrict__ maskcnt) {
    __shared__ float lds[8 * 768];                 // 8 waves * 48 rows * 16 cols
    const int lane = threadIdx.x & 31;
    const int wave = threadIdx.x >> 5;
    const int hi   = lane >> 4;                    // 0: K=0,1 / rows 0-7 ; 1: K=2,3 / rows 8-15
    const int n    = lane & 15;                    // column (point in tile) / A row
    const int p    = (blockIdx.x * 8 + wave) * 16 + n;

    // B fragment for this wave's 16 points
    float gx = grid[p*3 + 0], gy = grid[p*3 + 1], gz = grid[p*3 + 2];
    v2f b;
    b.x = hi ? gz : gx;
    b.y = hi ? 1.0f : gy;

    float* wl = &lds[wave * 768];
    v8f czero = {};
#pragma unroll
    for (int t = 0; t < 3; ++t) {
        const float* ap = wsA + (t*16 + n)*4 + hi*2;
        v2f a;
        a.x = ap[0];
        a.y = ap[1];
        v8f d = __builtin_amdgcn_wmma_f32_16x16x4_f32(
            false, a, false, b, (short)0, czero, false, false);
#pragma unroll
        for (int v = 0; v < 8; ++v)
            wl[(t*16 + v + hi*8) * 16 + n] = d[v];
    }
    __syncthreads();

    float mc = 0.0f;
#pragma unroll
    for (int j = 0; j < 3; ++j) {
        int c = 2*j + hi;                          // camera for this lane
        float pr0 = wl[(3*c + 0)*16 + n];
        float pr1 = wl[(3*c + 1)*16 + n];
        float pr2 = wl[(3*c + 2)*16 + n];          // == cam z (K row 2 = [0,0,1])
        float cr0 = wl[(18 + 3*c + 0)*16 + n];
        float cr1 = wl[(18 + 3*c + 1)*16 + n];
        float cr2 = wl[(18 + 3*c + 2)*16 + n];

        // mask sampling: clamp_z = False -> z + 1e-8
        float zm = pr2 + 1e-8f;
        int xm = clampi((int)rintf(pr0 / zm), 0, W - 1);
        int ym = clampi((int)rintf(pr1 / zm), 0, H - 1);
        mc += maskImg[c*HWPIX + ym*W + xm];

        // visibility / rgb coords: clamp_z = True -> max(z, 1e-8)
        float zc = fmaxf(pr2, 1e-8f);
        int xc = clampi((int)rintf(pr0 / zc), 0, W - 1);
        int yc = clampi((int)rintf(pr1 / zc), 0, H - 1);
        unsigned int pid = (unsigned int)(yc*W + xc);

        // ||p - cam_pos|| == ||R p + t|| (R orthonormal)
        float dist = sqrtf(cr0*cr0 + cr1*cr1 + cr2*cr2);

        distbuf[c*NPTS + p] = dist;
        pidbuf [c*NPTS + p] = pid;
    }
    // combine camera halves: lane n has cams 0,2,4; lane n+16 has cams 1,3,5
    mc += __shfl_xor(mc, 16, 32);
    if (hi == 0) maskcnt[p] = mc;                  // integer count 0..6 (exact in fp)
}

// ---------------------------------------------------------------------------
// Kernel 3: reset per-camera depth (+inf) and winner-index (NPTS) buffers.
// ---------------------------------------------------------------------------
__global__ void k_clear(unsigned int* __restrict__ zbuf,
                        unsigned int* __restrict__ argbuf) {
    int i = blockIdx.x * blockDim.x + threadIdx.x;
    if (i < NCAM * HWPIX) {
        zbuf[i]   = 0x7F800000u;                   // +inf bits
        argbuf[i] = (unsigned int)NPTS;
    }
}

// ---------------------------------------------------------------------------
// Kernel 4: segment_min of distance per pixel (atomicMin on uint bits; dist>=0
// so the bit pattern is monotone). Non-selected points are the inf case.
// ---------------------------------------------------------------------------
__global__ void k_depth(const float* __restrict__ distbuf,
                        const unsigned int* __restrict__ pidbuf,
                        const float* __restrict__ maskcnt,
                        unsigned int* __restrict__ zbuf,
                        float thresh) {
    int i = blockIdx.x * blockDim.x + threadIdx.x;
    if (i >= NCAM * NPTS) return;
    int c  = i / NPTS;
    int nn = i - c * NPTS;
    if (maskcnt[nn] < thresh) return;
    unsigned int bits = __float_as_uint(distbuf[i]);
    atomicMin(&zbuf[c*HWPIX + pidbuf[i]], bits);
}

// ---------------------------------------------------------------------------
// Kernel 5: tie-break — min point index among dist == segment min.
// ---------------------------------------------------------------------------
__global__ void k_winner(const float* __restrict__ distbuf,
                         const unsigned int* __restrict__ pidbuf,
                         const float* __restrict__ maskcnt,
                         const unsigned int* __restrict__ zbuf,
                         unsigned int* __restrict__ argbuf,
                         float thresh) {
    int i = blockIdx.x * blockDim.x + threadIdx.x;
    if (i >= NCAM * NPTS) return;
    int c  = i / NPTS;
    int nn = i - c * NPTS;
    if (maskcnt[nn] < thresh) return;
    unsigned int bits = __float_as_uint(distbuf[i]);
    unsigned int pid  = pidbuf[i];
    if (bits == zbuf[c*HWPIX + pid])
        atomicMin(&argbuf[c*HWPIX + pid], (unsigned int)nn);
}

// ---------------------------------------------------------------------------
// Kernel 6: per-point color blend + output accumulation (x0.5 per threshold).
// ---------------------------------------------------------------------------
__global__ void k_accum(const float* __restrict__ rgb,
                        const unsigned int* __restrict__ pidbuf,
                        const float* __restrict__ maskcnt,
                        const unsigned int* __restrict__ argbuf,
                        float* __restrict__ out,
                        float thresh, int first) {
    int nn = blockIdx.x * blockDim.x + threadIdx.x;
    if (nn >= NPTS) return;
    bool sel = maskcnt[nn] >= thresh;
    float o0, o1, o2, o3;
    if (sel) {
        float wsum = 0.0f, r = 0.0f, g = 0.0f, b = 0.0f;
        for (int c = 0; c < NCAM; ++c) {
            unsigned int pid = pidbuf[c*NPTS + nn];
            bool vis = (argbuf[c*HWPIX + pid] == (unsigned int)nn);
            float w = vis ? 1.0f : NVW;
            int y = (int)(pid >> 9), x = (int)(pid & 511u);
            const float* base = rgb + ((size_t)(c*3) * H + y) * W + x;
            r += w * base[0];
            g += w * base[HWPIX];
            b += w * base[2*HWPIX];
            wsum += w;
        }
        float inv = 1.0f / fmaxf(wsum, 1e-8f);
        o0 = 1.0f; o1 = r*inv; o2 = g*inv; o3 = b*inv;
    } else {
        o0 = 0.0f; o1 = FILLV; o2 = FILLV; o3 = FILLV;
    }
    if (first) {
        out[0*NPTS + nn] = 0.5f*o0;
        out[1*NPTS + nn] = 0.5f*o1;
        out[2*NPTS + nn] = 0.5f*o2;
        out[3*NPTS + nn] = 0.5f*o3;
    } else {
        out[0*NPTS + nn] += 0.5f*o0;
        out[1*NPTS + nn] += 0.5f*o1;
        out[2*NPTS + nn] += 0.5f*o2;
        out[3*NPTS + nn] += 0.5f*o3;
    }
}

// ---------------------------------------------------------------------------
extern "C" void kernel_launch(void* const* d_in, const int* in_sizes, int n_in,
                              void* d_out, int out_size, void* d_ws, size_t ws_size,
                              hipStream_t stream) {
    const float* mask   = (const float*)d_in[0];   // (6,1,512,512)
    const float* rgb    = (const float*)d_in[1];   // (6,3,512,512)
    const float* center = (const float*)d_in[2];   // (3,)
    const float* angle  = (const float*)d_in[3];   // (1,)
    const float* grid   = (const float*)d_in[4];   // (96,96,96,3)
    const float* Km     = (const float*)d_in[5];   // (6,3,3)
    const float* Em     = (const float*)d_in[6];   // (6,4,4)
    float* out = (float*)d_out;                    // (4,96,96,96)

    // workspace layout (4-byte elements): total ~58.6 MB
    float*        ws      = (float*)d_ws;
    float*        wsA     = ws;                                    // 256
    float*        distbuf = ws + 256;                              // 6*NPTS
    unsigned int* pidbuf  = (unsigned int*)(ws + 256 + NCAM*NPTS); // 6*NPTS
    float*        maskcnt = ws + 256 + 2*NCAM*NPTS;                // NPTS
    unsigned int* zbuf    = (unsigned int*)(ws + 256 + 2*NCAM*NPTS + NPTS);
    unsigned int* argbuf  = zbuf + NCAM*HWPIX;

    k_setup<<<1, 64, 0, stream>>>(center, angle, Km, Em, wsA);
    k_project<<<NPTS/128, 256, 0, stream>>>(grid, mask, wsA, distbuf, pidbuf, maskcnt);

    // thresholds: mean>=1.0 <=> count==6 ; mean>=5/6 <=> count>=5 (exact ints)
    const float thr[2] = {5.5f, 4.5f};
    for (int t = 0; t < 2; ++t) {
        k_clear <<<(NCAM*HWPIX + 255)/256, 256, 0, stream>>>(zbuf, argbuf);
        k_depth <<<(NCAM*NPTS + 255)/256, 256, 0, stream>>>(distbuf, pidbuf, maskcnt, zbuf, thr[t]);
        k_winner<<<(NCAM*NPTS + 255)/256, 256, 0, stream>>>(distbuf, pidbuf, maskcnt, zbuf, argbuf, thr[t]);
        k_accum <<<(NPTS + 255)/256, 256, 0, stream>>>(rgb, pidbuf, maskcnt, argbuf, out, thr[t], t == 0 ? 1 : 0);
    }
}